// BSplineBasis_25812753449578
// MI455X (gfx1250) — compile-verified
//
#include <hip/hip_runtime.h>

// B-spline basis (degree 3, 256 knots -> 252 basis funcs) for N=1M points.
// Store-bandwidth-bound: 1.057 GB of output, only 4 nonzeros per 252-wide row.
// Roofline: ~0.1 FLOP/byte -> HBM-store bound, floor ~45us at 23.3 TB/s.
// Strategy: build 64-row tiles in LDS (zeros + 4 sparse values/row), then
// stream to global with CDNA5 async LDS->global b128 stores (ASYNCcnt path),
// non-temporal so the 1 GB stream doesn't thrash the 192 MB L2.

#define NUM_KNOTS 256
#define NUM_BASIS 252
#define ROWS_PER_BLOCK 64
#define TPB 64  // 2 waves (wave32)

__global__ __launch_bounds__(TPB) void
bspline_basis_kernel(const float* __restrict__ x, const float* __restrict__ knots,
                     float* __restrict__ out, int n) {
    __shared__ __align__(16) float tile[ROWS_PER_BLOCK * NUM_BASIS];  // 64512 B
    const int tid  = (int)threadIdx.x;
    const int row0 = (int)blockIdx.x * ROWS_PER_BLOCK;
    const int nchunks = (ROWS_PER_BLOCK * NUM_BASIS) / 4;  // 4032 float4 chunks

    // ---- Phase A: zero the LDS tile with wide DS stores ----
    float4* t4 = (float4*)tile;
    #pragma unroll 4
    for (int i = tid; i < nchunks; i += TPB)
        t4[i] = make_float4(0.f, 0.f, 0.f, 0.f);
    __syncthreads();

    // ---- Phase B: one row per thread; local 4-wide Cox-de Boor ----
    const int row = row0 + tid;
    if (row < n) {
        const float xv   = x[row];
        const float kmin = knots[0];
        const float kmax = knots[NUM_KNOTS - 1];
        if (xv >= kmin && xv < kmax) {
            // Estimate interval from uniform spacing, then fix up against the
            // actual f32 linspace values (match reference's comparisons).
            int j = (int)((xv - kmin) * ((float)(NUM_KNOTS - 1) / (kmax - kmin)));
            if (j < 0) j = 0;
            if (j > NUM_KNOTS - 2) j = NUM_KNOTS - 2;
            while (j > 0 && xv < knots[j]) --j;
            while (j < NUM_KNOTS - 2 && xv >= knots[j + 1]) ++j;

            // knots[j-3 .. j+4], clamped; duplicate edge knots -> d==0 -> w=0
            float kn[8];
            #pragma unroll
            for (int t = 0; t < 8; ++t) {
                int ki = j - 3 + t;
                ki = ki < 0 ? 0 : (ki > NUM_KNOTS - 1 ? NUM_KNOTS - 1 : ki);
                kn[t] = knots[ki];
            }

            // P[t] = B_{j-3+t}^{k-1}; degree 0: only m=j is 1.
            float P[5] = {0.f, 0.f, 0.f, 1.f, 0.f};
            #pragma unroll
            for (int k = 1; k <= 3; ++k) {
                float Nv[4];
                #pragma unroll
                for (int t = 0; t < 4; ++t) {
                    const float d1 = kn[t + k] - kn[t];
                    const float d2 = kn[t + k + 1] - kn[t + 1];
                    const float w1 = (d1 > 0.f) ? (xv - kn[t]) / d1 : 0.f;
                    const float w2 = (d2 > 0.f) ? (kn[t + k + 1] - xv) / d2 : 0.f;
                    Nv[t] = w1 * P[t] + w2 * P[t + 1];
                }
                #pragma unroll
                for (int t = 0; t < 4; ++t) P[t] = Nv[t];
                P[4] = 0.f;
            }

            #pragma unroll
            for (int t = 0; t < 4; ++t) {
                const int m = j - 3 + t;
                if (m >= 0 && m < NUM_BASIS)
                    tile[tid * NUM_BASIS + m] = P[t];
            }
        }
    }
    __syncthreads();

    // ---- Phase C: stream tile -> global via CDNA5 async LDS->global stores ----
    // SADDR (GVS) form: SGPR pair holds the tile's global base, per-lane VGPR
    // carries a 32-bit byte offset shared with the LDS address computation.
    // th:TH_STORE_NT: write-once 1 GB stream, don't retain in caches.
    const unsigned lds_base = (unsigned)(unsigned long long)tile;  // low 32b = LDS offset
    const unsigned long long gbase =
        (unsigned long long)(out + (unsigned long long)row0 * NUM_BASIS);
    const int rows_here = (n - row0 < ROWS_PER_BLOCK) ? (n - row0) : ROWS_PER_BLOCK;
    const unsigned valid_bytes = (unsigned)(rows_here * NUM_BASIS) * 4u;
    #pragma unroll 2
    for (int i = tid; i < nchunks; i += TPB) {
        const unsigned boff = (unsigned)i * 16u;
        if (boff < valid_bytes) {
            const unsigned lds_addr = lds_base + boff;
            asm volatile("global_store_async_from_lds_b128 %0, %1, %2 th:TH_STORE_NT"
                         :: "v"(boff), "v"(lds_addr), "s"(gbase)
                         : "memory");
        }
    }
    asm volatile("s_wait_asynccnt 0x0" ::: "memory");
}

extern "C" void kernel_launch(void* const* d_in, const int* in_sizes, int n_in,
                              void* d_out, int out_size, void* d_ws, size_t ws_size,
                              hipStream_t stream) {
    const float* x     = (const float*)d_in[0];
    const float* knots = (const float*)d_in[1];
    float* out         = (float*)d_out;
    const int n        = in_sizes[0];  // 1048576
    const int blocks   = (n + ROWS_PER_BLOCK - 1) / ROWS_PER_BLOCK;
    bspline_basis_kernel<<<blocks, TPB, 0, stream>>>(x, knots, out, n);
}